// ECELoss_38139309588817
// MI455X (gfx1250) — compile-verified
//
#include <hip/hip_runtime.h>
#include <hip/hip_bf16.h>
#include <stdint.h>

// ECE loss, MI455X (gfx1250). Streaming-bandwidth bound: 512MB @ 23.3TB/s ~ 22us.
// Strategy: per-wave 3-deep ring of async global->LDS tiles (16 rows x 512B each,
// 2 tiles in flight per wave), XOR-swizzled LDS layout, 2 lanes per row, single
// shfl_xor combine, per-block LDS bin accumulators, tiny finalize kernel.

#define N_BINS 20
#define ROW_C 128            // classes per row (fixed by reference)
#define TILE_ROWS 16         // rows per tile per wave (8KB)
#define TILE_FLOATS (TILE_ROWS * ROW_C)
#define N_BUFS 3             // 3-stage ring: 2 tiles in flight while 1 is consumed
#define WAVES_PER_BLOCK 2    // 2 waves * 3 bufs * 8KB = 48KB LDS (+bins) per block
#define BLOCK_THREADS (WAVES_PER_BLOCK * 32)

#define AS1 __attribute__((address_space(1)))
#define AS3 __attribute__((address_space(3)))

typedef __attribute__((__vector_size__(4 * sizeof(int)))) int v4i;

// Non-temporal load hint: data is streamed exactly once (512MB > 192MB L2).
#define ASYNC_CPOL 1  // TH[2:0]=NT

__device__ __forceinline__ void async_copy_b128(const float* g, float* l) {
#if __has_builtin(__builtin_amdgcn_global_load_async_to_lds_b128)
  // CDNA5 async vmem->LDS (tracked by ASYNCcnt). Per-lane 16B copy.
  __builtin_amdgcn_global_load_async_to_lds_b128(
      (AS1 v4i*)(uintptr_t)g, (AS3 v4i*)(uintptr_t)l, 0, ASYNC_CPOL);
#else
  // Fallback: reg-staged copy; compiler inserts loadcnt/dscnt waits.
  *reinterpret_cast<float4*>(l) = *reinterpret_cast<const float4*>(g);
#endif
}

// Wait until at most `n` async-to-LDS ops remain outstanding. Each tile copy
// issues exactly 16 ops, so n = 16 * (#later tiles allowed in flight).
__device__ __forceinline__ void wait_async_le32() {
#if __has_builtin(__builtin_amdgcn_s_wait_asynccnt)
  __builtin_amdgcn_s_wait_asynccnt(32);
#else
  asm volatile("s_wait_asynccnt 0x20" ::: "memory");
#endif
  asm volatile("" ::: "memory");
}
__device__ __forceinline__ void wait_async_le16() {
#if __has_builtin(__builtin_amdgcn_s_wait_asynccnt)
  __builtin_amdgcn_s_wait_asynccnt(16);
#else
  asm volatile("s_wait_asynccnt 0x10" ::: "memory");
#endif
  asm volatile("" ::: "memory");
}
__device__ __forceinline__ void wait_async_le0() {
#if __has_builtin(__builtin_amdgcn_s_wait_asynccnt)
  __builtin_amdgcn_s_wait_asynccnt(0);
#else
  asm volatile("s_wait_asynccnt 0x0" ::: "memory");
#endif
  asm volatile("" ::: "memory");
}

// Copy one 16-row tile (rows rowBase..rowBase+15) into ldsBuf with XOR swizzle:
// chunk c (16B) of tile-row j lands at slot (c ^ j). Global side: each of the
// 16 async b128 instructions reads 512B contiguous (fully coalesced).
__device__ __forceinline__ void copy_tile_async(const float* __restrict__ src,
                                                long long rowBase, long long nRows,
                                                float* ldsBuf, int lane) {
#pragma unroll
  for (int j = 0; j < TILE_ROWS; ++j) {
    long long r = rowBase + j;
    if (r >= nRows) r = nRows - 1;  // clamp: keep issued-instruction count fixed
    const float* g = src + (size_t)r * ROW_C + ((size_t)lane << 2);
    int slot = lane ^ j;            // j < 16 so this permutes within 16-halves
    float* l = ldsBuf + (size_t)j * ROW_C + ((size_t)slot << 2);
    async_copy_b128(g, l);
  }
}

// Two lanes per row: lane = h*16 + r handles half h of tile-row r.
__device__ __forceinline__ void process_tile(const float* ldsBuf,
                                             long long rowBase, long long nRows,
                                             const int* __restrict__ labels,
                                             float* sbins, int lane) {
  const int r = lane & 15;
  const int h = lane >> 4;
  const float* rp = ldsBuf + (size_t)r * ROW_C;

  float m = -1.0f;
  int idx = 0;
#pragma unroll
  for (int i = 0; i < 16; ++i) {
    int s = (h << 4) + i;          // LDS slot we read (same slot in all 16 rows of this half)
    float4 v = *reinterpret_cast<const float4*>(rp + ((size_t)s << 2));
    int base = ((s ^ r) << 2);     // un-swizzle: actual element index base
    if (v.x > m) { m = v.x; idx = base; }
    if (v.y > m) { m = v.y; idx = base + 1; }
    if (v.z > m) { m = v.z; idx = base + 2; }
    if (v.w > m) { m = v.w; idx = base + 3; }
  }
  // Combine the two half-row partials (single butterfly step).
  float om = __shfl_xor(m, 16, 32);
  int   oi = __shfl_xor(idx, 16, 32);
  if (om > m || (om == m && oi < idx)) { m = om; idx = oi; }

  long long row = rowBase + r;
  if (h == 0 && row < nRows) {
    int lab = labels[row];
    float acc = (idx == lab) ? 1.0f : 0.0f;
    int b = (int)ceilf(m * (float)N_BINS) - 1;
    b = b < 0 ? 0 : (b > (N_BINS - 1) ? (N_BINS - 1) : b);
    atomicAdd(&sbins[b], m);             // ds_add_f32
    atomicAdd(&sbins[N_BINS + b], acc);  // ds_add_f32
  }
}

__global__ __launch_bounds__(BLOCK_THREADS)
void ece_main(const float* __restrict__ outputs, const int* __restrict__ labels,
              float* __restrict__ gbins, long long nRows, long long numTiles,
              long long totalWaves) {
  __shared__ __align__(16) float tiles[WAVES_PER_BLOCK][N_BUFS][TILE_FLOATS];
  __shared__ float sbins[2 * N_BINS];

  for (int i = threadIdx.x; i < 2 * N_BINS; i += blockDim.x) sbins[i] = 0.0f;
  __syncthreads();

  const int lane = threadIdx.x & 31;
  const int w = threadIdx.x >> 5;
  const long long waveId = (long long)blockIdx.x * WAVES_PER_BLOCK + w;

  // Prologue: prime two pipeline stages.
  if (waveId < numTiles)
    copy_tile_async(outputs, waveId * TILE_ROWS, nRows, &tiles[w][0][0], lane);
  if (waveId + totalWaves < numTiles)
    copy_tile_async(outputs, (waveId + totalWaves) * TILE_ROWS, nRows,
                    &tiles[w][1][0], lane);

  int cur = 0;
  for (long long t = waveId; t < numTiles; t += totalWaves) {
    long long tn1 = t + totalWaves;
    long long tn2 = t + 2 * totalWaves;
    if (tn2 < numTiles) {
      int nb = cur + 2; if (nb >= N_BUFS) nb -= N_BUFS;
      copy_tile_async(outputs, tn2 * TILE_ROWS, nRows, &tiles[w][nb][0], lane);
      wait_async_le32();        // tiles tn1, tn2 may remain in flight
    } else if (tn1 < numTiles) {
      wait_async_le16();        // tile tn1 may remain in flight
    } else {
      wait_async_le0();         // last tile: drain
    }
    process_tile(&tiles[w][cur][0], t * TILE_ROWS, nRows, labels, sbins, lane);
    cur += 1; if (cur >= N_BUFS) cur -= N_BUFS;
  }

  __syncthreads();
  for (int i = threadIdx.x; i < 2 * N_BINS; i += blockDim.x)
    atomicAdd(&gbins[i], sbins[i]);      // global_atomic_add_f32, 40/block
}

__global__ void ece_zero(float* gbins) {
  int i = threadIdx.x;
  if (i < 2 * N_BINS) gbins[i] = 0.0f;
}

__global__ void ece_final(const float* __restrict__ gbins, float* __restrict__ out,
                          float invN) {
  int i = threadIdx.x;  // one wave
  float d = (i < N_BINS) ? fabsf(gbins[i] - gbins[N_BINS + i]) : 0.0f;
#pragma unroll
  for (int off = 16; off > 0; off >>= 1) d += __shfl_xor(d, off, 32);
  if (i == 0) out[0] = d * invN;
}

extern "C" void kernel_launch(void* const* d_in, const int* in_sizes, int n_in,
                              void* d_out, int out_size, void* d_ws, size_t ws_size,
                              hipStream_t stream) {
  const float* outputs = (const float*)d_in[0];
  const int* labels = (const int*)d_in[1];
  const long long nRows = (long long)in_sizes[1];  // labels count == N
  float* gbins = (float*)d_ws;                     // [0..19]=sum_conf, [20..39]=sum_acc

  const long long numTiles = (nRows + TILE_ROWS - 1) / TILE_ROWS;
  long long wantBlocks = (numTiles + WAVES_PER_BLOCK - 1) / WAVES_PER_BLOCK;
  int blocks = (int)(wantBlocks < 2048 ? (wantBlocks > 0 ? wantBlocks : 1) : 2048);
  const long long totalWaves = (long long)blocks * WAVES_PER_BLOCK;

  ece_zero<<<1, 64, 0, stream>>>(gbins);
  ece_main<<<blocks, BLOCK_THREADS, 0, stream>>>(outputs, labels, gbins, nRows,
                                                 numTiles, totalWaves);
  ece_final<<<1, 32, 0, stream>>>(gbins, (float*)d_out, 1.0f / (float)nRows);
}